// NISER_ODE_58746562674833
// MI455X (gfx1250) — compile-verified
//
#include <hip/hip_runtime.h>
#include <math.h>

// ---------------------------------------------------------------- constants
#define Vv    50000
#define Dd    128
#define Bb    1024
#define Pp    50
#define Nn    (Bb*Pp)     // 51200 nodes
#define Ee    (Bb*Pp)     // 51200 edges
#define D2    256
#define D3    384
#define D4    512
#define SCALEF 12.0f

typedef __attribute__((ext_vector_type(2))) float v2f;
typedef __attribute__((ext_vector_type(8))) float v8f;

__device__ __forceinline__ float sigf(float x) { return 1.0f / (1.0f + expf(-x)); }

// ---------------------------------------------------------------- WMMA GEMM
// C[M x Ncol] = (RS? diag(rowscale):I) * A[M x K] @ B  (+bias) (+Cin)
// A row-major (lda); B row-major (ldb); TRANSB: element(k,n) = Bm[n*ldb + k].
// Each wave computes a 64 x (16*NT) block: 4 stacked M-tiles x NT N-tiles.
// Per K-step: 4 A-fragment loads + NT B-fragment loads feed 4*NT WMMAs.
// Fragments are double-buffered (load kb+4 while multiplying kb) so WMMAs
// overlap load latency. Rowscale is applied in the epilogue (linearity).
// Requires M%64==0, Ncol%(16*NT)==0, K%4==0.
template<bool TRANSB, int NT>
__device__ __forceinline__ void gemm_load_frags(const float* const* Arow,
                                                const float* const* Bcol,
                                                int ldb, int k0, v2f* a, v2f* b)
{
#pragma unroll
    for (int r = 0; r < 4; ++r) {
        a[r].x = Arow[r][k0];
        a[r].y = Arow[r][k0 + 1];
    }
#pragma unroll
    for (int j = 0; j < NT; ++j) {
        if (TRANSB) {                          // consecutive k: one b64 load
            b[j].x = Bcol[j][k0];
            b[j].y = Bcol[j][k0 + 1];
        } else {                               // strided rows: two b32 loads
            b[j].x = Bcol[j][(size_t)k0 * ldb];
            b[j].y = Bcol[j][(size_t)(k0 + 1) * ldb];
        }
    }
}

template<bool TRANSB, bool BIAS, bool RS, bool CIN, int NT>
__global__ void k_gemm(const float* __restrict__ A, int lda,
                       const float* __restrict__ Bm, int ldb,
                       const float* __restrict__ bias,
                       const float* __restrict__ rowscale,
                       const float* Cin, float* C, int ldc,
                       int M, int Ncol, int K)
{
    int wave   = blockIdx.x * (blockDim.x >> 5) + (threadIdx.x >> 5);
    int tilesN = Ncol / (16 * NT);
    int tilesM = M >> 6;                       // 64-row strips
    if (wave >= tilesM * tilesN) return;
    int tm = wave / tilesN, tn = wave - tm * tilesN;
    int lane = threadIdx.x & 31;
    int half = lane >> 4;                      // 0 -> K pair {0,1}; 1 -> {2,3}
    int lrow = lane & 15;
    int bcol0 = tn * 16 * NT + lrow;

    const float* Arow[4];
#pragma unroll
    for (int r = 0; r < 4; ++r)
        Arow[r] = A + (size_t)(tm * 64 + 16 * r + lrow) * lda;
    const float* Bcol[NT];
#pragma unroll
    for (int j = 0; j < NT; ++j)
        Bcol[j] = TRANSB ? (Bm + (size_t)(bcol0 + 16 * j) * ldb)
                         : (Bm + bcol0 + 16 * j);

    v8f zero = {};
    v8f acc[4][NT];
#pragma unroll
    for (int r = 0; r < 4; ++r)
#pragma unroll
        for (int j = 0; j < NT; ++j) acc[r][j] = zero;

    v2f aC[4], bC[NT];
    gemm_load_frags<TRANSB, NT>(Arow, Bcol, ldb, 2 * half, aC, bC);

    for (int kb = 0; kb + 4 < K; kb += 4) {
        v2f aN[4], bN[NT];
        gemm_load_frags<TRANSB, NT>(Arow, Bcol, ldb, kb + 4 + 2 * half, aN, bN);
        __builtin_prefetch(Arow[0] + kb + 64, 0, 0);   // global_prefetch_b8
#pragma unroll
        for (int r = 0; r < 4; ++r)
#pragma unroll
            for (int j = 0; j < NT; ++j)
                // (neg_a, A, neg_b, B, c_mod, C, reuse_a, reuse_b)
                acc[r][j] = __builtin_amdgcn_wmma_f32_16x16x4_f32(
                    false, aC[r], false, bC[j], (short)0, acc[r][j], false, false);
#pragma unroll
        for (int r = 0; r < 4; ++r) aC[r] = aN[r];
#pragma unroll
        for (int j = 0; j < NT; ++j) bC[j] = bN[j];
    }
#pragma unroll
    for (int r = 0; r < 4; ++r)
#pragma unroll
        for (int j = 0; j < NT; ++j)
            acc[r][j] = __builtin_amdgcn_wmma_f32_16x16x4_f32(
                false, aC[r], false, bC[j], (short)0, acc[r][j], false, false);

    float bv[NT];
#pragma unroll
    for (int j = 0; j < NT; ++j) bv[j] = BIAS ? bias[bcol0 + 16 * j] : 0.0f;
#pragma unroll
    for (int r = 0; r < 4; ++r) {
#pragma unroll
        for (int rr = 0; rr < 8; ++rr) {
            int mrow = tm * 64 + 16 * r + rr + 8 * half; // lanes>=16 hold M=r+8
            float s = RS ? rowscale[mrow] : 1.0f;
#pragma unroll
            for (int j = 0; j < NT; ++j) {
                int bcol = bcol0 + 16 * j;
                float v = acc[r][j][rr] * s + bv[j];
                if (CIN) v += Cin[(size_t)mrow * ldc + bcol];
                C[(size_t)mrow * ldc + bcol] = v;
            }
        }
    }
}

template<int NT>
static inline void launch_gemm_nt(hipStream_t stream,
                                  const float* A, int lda, const float* Bm, int ldb,
                                  bool transB, const float* bias, const float* rs,
                                  const float* Cin, float* C, int ldc,
                                  int M, int Ncol, int K)
{
    int tiles  = (M >> 6) * (Ncol / (16 * NT));
    int blocks = (tiles + 7) / 8;               // 8 waves per 256-thread block
#define GEMM_GO(TB, BI, RSF, CI) \
    k_gemm<TB, BI, RSF, CI, NT><<<blocks, 256, 0, stream>>>(A, lda, Bm, ldb, bias, rs, \
                                                            Cin, C, ldc, M, Ncol, K)
    if (transB) { if (bias) GEMM_GO(true, true, false, false);
                  else      GEMM_GO(true, false, false, false); }
    else if (Cin)  GEMM_GO(false, false, false, true);
    else if (rs)   GEMM_GO(false, true, true, false);   // rs call sites all have bias
    else if (bias) GEMM_GO(false, true, false, false);
    else           GEMM_GO(false, false, false, false);
#undef GEMM_GO
}

static inline void launch_gemm(hipStream_t stream,
                               const float* A, int lda, const float* Bm, int ldb,
                               bool transB, const float* bias, const float* rs,
                               const float* Cin, float* C, int ldc,
                               int M, int Ncol, int K)
{
    int colMain = (Ncol % 32 == 0) ? Ncol : Ncol - 16;   // all Ncol here are %16==0
    if (colMain > 0)
        launch_gemm_nt<2>(stream, A, lda, Bm, ldb, transB, bias, rs, Cin,
                          C, ldc, M, colMain, K);
    if (colMain != Ncol) {                                // 16-column tail (logits GEMM)
        int off = colMain;
        const float* Bt = transB ? (Bm + (size_t)off * ldb) : (Bm + off);
        launch_gemm_nt<1>(stream, A, lda, Bt, ldb, transB,
                          bias ? bias + off : nullptr, rs,
                          Cin ? Cin + off : nullptr, C + off, ldc,
                          M, Ncol - colMain, K);
    }
}

// ---------------------------------------------------------------- utilities
__global__ void k_maxpos(const float* __restrict__ v, int n, float* out) {
    __shared__ float sh[256];
    float m = 0.0f;
    for (int i = blockIdx.x * blockDim.x + threadIdx.x; i < n; i += gridDim.x * blockDim.x)
        m = fmaxf(m, v[i]);
    sh[threadIdx.x] = m; __syncthreads();
    for (int s = 128; s > 0; s >>= 1) {
        if (threadIdx.x < s) sh[threadIdx.x] = fmaxf(sh[threadIdx.x], sh[threadIdx.x + s]);
        __syncthreads();
    }
    if (threadIdx.x == 0) atomicMax((unsigned int*)out, __float_as_uint(sh[0]));
}

// dst[row] = src[idx?idx[row]:row] / (||.|| + 1e-12)
__global__ void k_rownorm_gather(const float* __restrict__ src, const int* __restrict__ idx,
                                 float* __restrict__ dst) {
    __shared__ float sh[Dd];
    int row = blockIdx.x, d = threadIdx.x;
    int sr = idx ? idx[row] : row;
    float v = src[(size_t)sr * Dd + d];
    sh[d] = v * v; __syncthreads();
    for (int s = 64; s > 0; s >>= 1) { if (d < s) sh[d] += sh[d + s]; __syncthreads(); }
    dst[(size_t)row * Dd + d] = v / (sqrtf(sh[0]) + 1e-12f);
}

// in-place row normalize; mode 0: /(n+eps)  mode 1: /max(n,eps)
__global__ void k_rownorm(float* __restrict__ x, int mode, float eps) {
    __shared__ float sh[Dd];
    int row = blockIdx.x, d = threadIdx.x;
    size_t i = (size_t)row * Dd + d;
    float v = x[i];
    sh[d] = v * v; __syncthreads();
    for (int s = 64; s > 0; s >>= 1) { if (d < s) sh[d] += sh[d + s]; __syncthreads(); }
    float nr = sqrtf(sh[0]);
    x[i] = v / (mode ? fmaxf(nr, eps) : (nr + eps));
}

// ---------------------------------------------------------------- phase 1
__global__ void k_ws_scatter(const int* __restrict__ src, const int* __restrict__ dst,
                             const float* __restrict__ w, float* wsin, float* wsout, int E) {
    int e = blockIdx.x * blockDim.x + threadIdx.x;
    if (e >= E) return;
    atomicAdd(&wsin[dst[e]], w[e]);
    atomicAdd(&wsout[src[e]], w[e]);
}

__global__ void k_msg_scatter(const int* __restrict__ src, const int* __restrict__ dst,
                              const float* __restrict__ w, const float* __restrict__ feat,
                              float* min_, float* mout) {
    int e = blockIdx.x, t = threadIdx.x;
    int s = src[e], d = dst[e];
    float we = w[e];
    atomicAdd(&min_[(size_t)d * Dd + t], feat[(size_t)s * Dd + t] * we);
    atomicAdd(&mout[(size_t)s * Dd + t], feat[(size_t)d * Dd + t] * we);
}

__global__ void k_div_ws(float* __restrict__ m, const float* __restrict__ wsum, int total) {
    int i = blockIdx.x * blockDim.x + threadIdx.x;
    if (i >= total) return;
    float w = wsum[i >> 7];
    m[i] /= (w > 0.0f ? w : 1.0f);
}

// GRU update + renorm (max(norm,1e-12))
__global__ void k_gru(float* __restrict__ feat, const float* __restrict__ gi,
                      const float* __restrict__ gh) {
    __shared__ float sh[Dd];
    int row = blockIdx.x, d = threadIdx.x;
    size_t b3 = (size_t)row * D3;
    float r = sigf(gi[b3 + d] + gh[b3 + d]);
    float z = sigf(gi[b3 + Dd + d] + gh[b3 + Dd + d]);
    float n = tanhf(gi[b3 + 2 * Dd + d] + r * gh[b3 + 2 * Dd + d]);
    size_t i = (size_t)row * Dd + d;
    float f = (1.0f - z) * n + z * feat[i];
    sh[d] = f * f; __syncthreads();
    for (int s = 64; s > 0; s >>= 1) { if (d < s) sh[d] += sh[d + s]; __syncthreads(); }
    feat[i] = f / fmaxf(sqrtf(sh[0]), 1e-12f);
}

// ---------------------------------------------------------------- ODE / GCN
__global__ void k_em_deg(const int* __restrict__ src, const int* __restrict__ dst,
                         const float* __restrict__ et, const float* __restrict__ dtp,
                         float tvmul, float* deg, int E) {
    int e = blockIdx.x * blockDim.x + threadIdx.x;
    if (e >= E) return;
    float tv = tvmul * dtp[0];
    int s = src[e], d = dst[e];
    if (et[e] <= tv && s != d) { atomicAdd(&deg[s], 1.0f); atomicAdd(&deg[d], 1.0f); }
}

__global__ void k_invsqrt_deg(const float* __restrict__ deg, float* __restrict__ nrm, int n) {
    int i = blockIdx.x * blockDim.x + threadIdx.x;
    if (i >= n) return;
    nrm[i] = rsqrtf(fmaxf(deg[i], 1.0f));
}

__global__ void k_gcn_scatter(const int* __restrict__ src, const int* __restrict__ dst,
                              const float* __restrict__ et, const float* __restrict__ dtp,
                              float tvmul, const float* __restrict__ nrm,
                              const float* __restrict__ v, float* agg) {
    int e = blockIdx.x;
    float tv = tvmul * dtp[0];
    int s = src[e], d = dst[e];
    if (!(et[e] <= tv && s != d)) return;
    int t = threadIdx.x;
    atomicAdd(&agg[(size_t)d * Dd + t], v[(size_t)s * Dd + t] * nrm[s]);
    atomicAdd(&agg[(size_t)s * Dd + t], v[(size_t)d * Dd + t] * nrm[d]);
}

__global__ void k_rz_rh(float* __restrict__ xr, const float* __restrict__ hr,
                        float* __restrict__ xz, const float* __restrict__ hz,
                        const float* __restrict__ h, float* __restrict__ rh, int n) {
    int i = blockIdx.x * blockDim.x + threadIdx.x;
    if (i >= n) return;
    float r = sigf(xr[i] + hr[i]);
    float z = sigf(xz[i] + hz[i]);
    xr[i] = r; xz[i] = z; rh[i] = r * h[i];
}

__global__ void k_unorm_k(const float* __restrict__ xh, const float* __restrict__ up,
                          const float* __restrict__ z, const float* __restrict__ h,
                          float* __restrict__ k) {
    __shared__ float sh[Dd];
    int row = blockIdx.x, d = threadIdx.x;
    size_t i = (size_t)row * Dd + d;
    float u = tanhf(xh[i] + up[i]);
    float dh = (1.0f - z[i]) * (u - h[i]);
    sh[d] = dh * dh; __syncthreads();
    for (int s = 64; s > 0; s >>= 1) { if (d < s) sh[d] += sh[d + s]; __syncthreads(); }
    k[i] = dh / fmaxf(sqrtf(sh[0]), 1e-12f);
}

__global__ void k_stage_update(float* __restrict__ kacc, const float* __restrict__ kcur,
                               float w, float* __restrict__ h, const float* __restrict__ hbase,
                               float c, const float* __restrict__ dtp, int n) {
    int i = blockIdx.x * blockDim.x + threadIdx.x;
    if (i >= n) return;
    float k = kcur[i];
    kacc[i] += w * k;
    if (c != 0.0f) h[i] = hbase[i] + c * dtp[0] * k;
}

__global__ void k_final_comb(float* __restrict__ x, const float* __restrict__ hbase,
                             const float* __restrict__ kacc, const float* __restrict__ dtp, int n) {
    int i = blockIdx.x * blockDim.x + threadIdx.x;
    if (i >= n) return;
    x[i] = hbase[i] + dtp[0] * (1.0f / 6.0f) * kacc[i];
}

// ---------------------------------------------------------------- attention
__global__ void k_gather_rows(const float* __restrict__ feat, const int* __restrict__ idx,
                              float* __restrict__ out) {
    int b = blockIdx.x, d = threadIdx.x;
    out[(size_t)b * Dd + d] = feat[(size_t)idx[b] * Dd + d];
}

__global__ void k_attn_e(const float* __restrict__ fu, const float* __restrict__ fv,
                         const int* __restrict__ gid, const float* __restrict__ fce,
                         float* __restrict__ e) {
    __shared__ float sh[Dd];
    int row = blockIdx.x, d = threadIdx.x;
    int g = gid[row];
    sh[d] = sigf(fu[(size_t)row * Dd + d] + fv[(size_t)g * Dd + d]) * fce[d];
    __syncthreads();
    for (int s = 64; s > 0; s >>= 1) { if (d < s) sh[d] += sh[d + s]; __syncthreads(); }
    if (d == 0) e[row] = sh[0];
}

// per-session softmax + weighted readout; segments are contiguous (P=50)
__global__ void k_session(const float* __restrict__ e, const float* __restrict__ feat,
                          const float* __restrict__ lastf, float* __restrict__ srcat) {
    __shared__ float sh[Dd];
    __shared__ float al[64];
    int b = blockIdx.x, t = threadIdx.x;
    sh[t] = (t < Pp) ? e[b * Pp + t] : -3.4e38f;
    __syncthreads();
    for (int s = 64; s > 0; s >>= 1) { if (t < s) sh[t] = fmaxf(sh[t], sh[t + s]); __syncthreads(); }
    float m = sh[0]; __syncthreads();
    float ex = (t < Pp) ? expf(e[b * Pp + t] - m) : 0.0f;
    sh[t] = ex; __syncthreads();
    for (int s = 64; s > 0; s >>= 1) { if (t < s) sh[t] += sh[t + s]; __syncthreads(); }
    float sum = sh[0];
    if (t < Pp) al[t] = ex / sum;
    __syncthreads();
    float acc = 0.0f;
    for (int j = 0; j < Pp; ++j)
        acc += feat[((size_t)(b * Pp + j)) * Dd + t] * al[j];
    srcat[(size_t)b * D2 + t]      = lastf[(size_t)b * Dd + t];
    srcat[(size_t)b * D2 + Dd + t] = acc;
}

// in-place log_softmax(scale * row) over V columns
__global__ void k_logsoftmax(float* __restrict__ out, int Vn, float scale) {
    __shared__ float sh[256];
    size_t base = (size_t)blockIdx.x * Vn;
    int t = threadIdx.x;
    float m = -3.4e38f;
    for (int i = t; i < Vn; i += 256) m = fmaxf(m, scale * out[base + i]);
    sh[t] = m; __syncthreads();
    for (int s = 128; s > 0; s >>= 1) { if (t < s) sh[t] = fmaxf(sh[t], sh[t + s]); __syncthreads(); }
    m = sh[0]; __syncthreads();
    float sum = 0.0f;
    for (int i = t; i < Vn; i += 256) sum += expf(scale * out[base + i] - m);
    sh[t] = sum; __syncthreads();
    for (int s = 128; s > 0; s >>= 1) { if (t < s) sh[t] += sh[t + s]; __syncthreads(); }
    float l = logf(sh[0]);
    for (int i = t; i < Vn; i += 256) out[base + i] = scale * out[base + i] - m - l;
}

// ---------------------------------------------------------------- launcher
extern "C" void kernel_launch(void* const* d_in, const int* in_sizes, int n_in,
                              void* d_out, int out_size, void* d_ws, size_t ws_size,
                              hipStream_t stream) {
    (void)in_sizes; (void)n_in; (void)out_size; (void)ws_size;
    const int*   iid  = (const int*)d_in[0];
    const int*   esrc = (const int*)d_in[1];
    const int*   edst = (const int*)d_in[2];
    const int*   gid  = (const int*)d_in[3];
    const int*   last = (const int*)d_in[4];
    const float* ew   = (const float*)d_in[5];
    const float* et   = (const float*)d_in[6];
    const float* emb  = (const float*)d_in[7];
    const float* W1   = (const float*)d_in[8];
    const float* W2   = (const float*)d_in[9];
    const float* wih  = (const float*)d_in[10];
    const float* whh  = (const float*)d_in[11];
    const float* bih  = (const float*)d_in[12];
    const float* bhh  = (const float*)d_in[13];
    const float* Wxr  = (const float*)d_in[14];
    const float* bxr  = (const float*)d_in[15];
    const float* Wxz  = (const float*)d_in[16];
    const float* bxz  = (const float*)d_in[17];
    const float* Wxh  = (const float*)d_in[18];
    const float* bxh  = (const float*)d_in[19];
    const float* Whr  = (const float*)d_in[20];
    const float* bhr  = (const float*)d_in[21];
    const float* Whz  = (const float*)d_in[22];
    const float* bhz  = (const float*)d_in[23];
    const float* Whh2 = (const float*)d_in[24];
    const float* bhh2 = (const float*)d_in[25];
    const float* fc_u = (const float*)d_in[26];
    const float* fc_vw= (const float*)d_in[27];
    const float* fc_vb= (const float*)d_in[28];
    const float* fc_e = (const float*)d_in[29];
    const float* fc_sr= (const float*)d_in[30];
    float* out = (float*)d_out;

    // ------- workspace layout (12 N*D slots + small region)
    const size_t ND = (size_t)Nn * Dd;
    float* ws = (float*)d_ws;
    float* x     = ws + 0  * ND;   // feat ("x" in ODE)
    float* hbase = ws + 1  * ND;
    float* kacc  = ws + 2  * ND;   // pre-ODE alias: gh (slots 2..4, N x 3D)
    float* h     = ws + 3  * ND;
    float* kcur  = ws + 4  * ND;
    float* t1    = ws + 5  * ND;   // phase1: m_in / neigh1
    float* t2    = ws + 6  * ND;   // phase1: m_out / neigh2
    float* xr    = ws + 7  * ND;   // pre-ODE alias: gi (slots 7..9, N x 3D)
    float* xz    = ws + 8  * ND;
    float* xh    = ws + 9  * ND;
    float* rh    = ws + 10 * ND;
    float* target= ws + 11 * ND;   // V*D = 6.40M < ND
    float* sm    = ws + 12 * ND;
    float* ws_in  = sm;
    float* ws_out = ws_in  + Nn;
    float* deg    = ws_out + Nn;
    float* nrm    = deg    + Nn;
    float* ebuf   = nrm    + Nn;
    float* fv     = ebuf   + Nn;
    float* lastf  = fv     + (size_t)Bb * Dd;
    float* srcat  = lastf  + (size_t)Bb * Dd;
    float* srb    = srcat  + (size_t)Bb * D2;
    float* C1     = srb    + (size_t)Bb * Dd;
    float* C2     = C1     + (size_t)Dd * D3;
    float* dtp    = C2     + (size_t)Dd * D3;
    float* gi = xr;     // N x 384
    float* gh = kacc;   // N x 384 (used only before the ODE begins)

    auto gemm = [&](const float* A, int lda, const float* Bm, int ldb, bool transB,
                    const float* bias, const float* rs, const float* Cin,
                    float* C, int ldc, int M, int Ncol, int K) {
        launch_gemm(stream, A, lda, Bm, ldb, transB, bias, rs, Cin, C, ldc, M, Ncol, K);
    };
    const int ndInt = (int)ND;

    // ------- dt = max(edge_t)
    hipMemsetAsync(dtp, 0, sizeof(float), stream);
    k_maxpos<<<256, 256, 0, stream>>>(et, Ee, dtp);

    // ------- phase 1: embed + normalize; normalized target embedding
    k_rownorm_gather<<<Nn, Dd, 0, stream>>>(emb, iid, x);
    k_rownorm_gather<<<Vv, Dd, 0, stream>>>(emb, nullptr, target);

    // ------- weighted neighbor aggregation
    hipMemsetAsync(ws_in, 0, 2 * Nn * sizeof(float), stream);
    hipMemsetAsync(t1, 0, ND * sizeof(float), stream);
    hipMemsetAsync(t2, 0, ND * sizeof(float), stream);
    k_ws_scatter<<<(Ee + 255) / 256, 256, 0, stream>>>(esrc, edst, ew, ws_in, ws_out, Ee);
    k_msg_scatter<<<Ee, Dd, 0, stream>>>(esrc, edst, ew, x, t1, t2);
    k_div_ws<<<(ndInt + 255) / 256, 256, 0, stream>>>(t1, ws_in, ndInt);
    k_div_ws<<<(ndInt + 255) / 256, 256, 0, stream>>>(t2, ws_out, ndInt);

    // ------- combined GRU-input matrices: C1 = W1 @ Wih[:, :2D].T, C2 = W2 @ Wih[:, 2D:].T
    gemm(W1, D2, wih,      D4, true, nullptr, nullptr, nullptr, C1, D3, Dd, D3, D2);
    gemm(W2, D2, wih + D2, D4, true, nullptr, nullptr, nullptr, C2, D3, Dd, D3, D2);

    // ------- gi = n1@C1 + n2@C2 + bih ; gh = feat@whh.T + bhh ; GRU
    gemm(t1, Dd, C1, D3, false, bih, nullptr, nullptr, gi, D3, Nn, D3, Dd);
    gemm(t2, Dd, C2, D3, false, nullptr, nullptr, gi,   gi, D3, Nn, D3, Dd);
    gemm(x,  Dd, whh, Dd, true, bhh, nullptr, nullptr, gh, D3, Nn, D3, Dd);
    k_gru<<<Nn, Dd, 0, stream>>>(x, gi, gh);

    // ------- ODE setup: hbase = h = x ; kacc = 0
    hipMemcpyAsync(hbase, x, ND * sizeof(float), hipMemcpyDeviceToDevice, stream);
    hipMemcpyAsync(h,     x, ND * sizeof(float), hipMemcpyDeviceToDevice, stream);
    hipMemsetAsync(kacc, 0, ND * sizeof(float), stream);

    auto stage = [&](float tvmul, float wK, float cnext) {
        hipMemsetAsync(deg, 0, Nn * sizeof(float), stream);
        k_em_deg<<<(Ee + 255) / 256, 256, 0, stream>>>(esrc, edst, et, dtp, tvmul, deg, Ee);
        k_invsqrt_deg<<<(Nn + 255) / 256, 256, 0, stream>>>(deg, nrm, Nn);
        // aggX -> xr/xz/xh
        hipMemsetAsync(t1, 0, ND * sizeof(float), stream);
        k_gcn_scatter<<<Ee, Dd, 0, stream>>>(esrc, edst, et, dtp, tvmul, nrm, x, t1);
        gemm(t1, Dd, Wxr, Dd, false, bxr, nrm, nullptr, xr, Dd, Nn, Dd, Dd);
        gemm(t1, Dd, Wxz, Dd, false, bxz, nrm, nullptr, xz, Dd, Nn, Dd, Dd);
        gemm(t1, Dd, Wxh, Dd, false, bxh, nrm, nullptr, xh, Dd, Nn, Dd, Dd);
        // aggH -> hr_lin(t1), hz_lin(kcur) ; r,z,rh
        hipMemsetAsync(t2, 0, ND * sizeof(float), stream);
        k_gcn_scatter<<<Ee, Dd, 0, stream>>>(esrc, edst, et, dtp, tvmul, nrm, h, t2);
        gemm(t2, Dd, Whr, Dd, false, bhr, nrm, nullptr, t1,   Dd, Nn, Dd, Dd);
        gemm(t2, Dd, Whz, Dd, false, bhz, nrm, nullptr, kcur, Dd, Nn, Dd, Dd);
        k_rz_rh<<<(ndInt + 255) / 256, 256, 0, stream>>>(xr, t1, xz, kcur, h, rh, ndInt);
        // agg(r*h) -> u_pre(t1) ; k = normalize((1-z)(tanh(xh+u_pre)-h))
        hipMemsetAsync(t2, 0, ND * sizeof(float), stream);
        k_gcn_scatter<<<Ee, Dd, 0, stream>>>(esrc, edst, et, dtp, tvmul, nrm, rh, t2);
        gemm(t2, Dd, Whh2, Dd, false, bhh2, nrm, nullptr, t1, Dd, Nn, Dd, Dd);
        k_unorm_k<<<Nn, Dd, 0, stream>>>(xh, t1, xz, h, kcur);
        k_stage_update<<<(ndInt + 255) / 256, 256, 0, stream>>>(kacc, kcur, wK, h, hbase,
                                                                cnext, dtp, ndInt);
    };
    stage(0.0f, 1.0f, 0.5f);   // k1
    stage(0.5f, 2.0f, 0.5f);   // k2
    stage(0.5f, 2.0f, 1.0f);   // k3
    stage(1.0f, 1.0f, 0.0f);   // k4

    // feat = normalize(hbase + dt/6 * kacc)
    k_final_comb<<<(ndInt + 255) / 256, 256, 0, stream>>>(x, hbase, kacc, dtp, ndInt);
    k_rownorm<<<Nn, Dd, 0, stream>>>(x, 1, 0.0f);

    // ------- attention readout
    gemm(x, Dd, fc_u, Dd, false, nullptr, nullptr, nullptr, t1, Dd, Nn, Dd, Dd);   // fu
    k_gather_rows<<<Bb, Dd, 0, stream>>>(x, last, lastf);
    gemm(lastf, Dd, fc_vw, Dd, false, fc_vb, nullptr, nullptr, fv, Dd, Bb, Dd, Dd);
    k_attn_e<<<Nn, Dd, 0, stream>>>(t1, fv, gid, fc_e, ebuf);
    k_session<<<Bb, Dd, 0, stream>>>(ebuf, x, lastf, srcat);
    gemm(srcat, D2, fc_sr, Dd, false, nullptr, nullptr, nullptr, srb, Dd, Bb, Dd, D2);
    k_rownorm<<<Bb, Dd, 0, stream>>>(srb, 0, 1e-12f);

    // ------- logits = sr @ target.T ; log_softmax(12 * logits)
    gemm(srb, Dd, target, Dd, true, nullptr, nullptr, nullptr, out, Vv, Bb, Vv, Dd);
    k_logsoftmax<<<Bb, 256, 0, stream>>>(out, Vv, SCALEF);
}